// ARModel_27925877359180
// MI455X (gfx1250) — compile-verified
//
#include <hip/hip_runtime.h>

// ============================================================================
// AR(16), blocked 16 steps/group:  W_{g+1} = M·W_g + L·E_g   (all 16x16),
// M = A^16 (companion), L[:,j] = std·(A^(15-j))[:,15].  One wave = 16 batch
// rows; each group = 8x V_WMMA_F32_16X16X4_F32, with D->B operand recycling
// done by lane-half swaps (ds_bpermute), no LDS round trip.
//
// Occupancy fix (serial chain is the bottleneck, not HBM):  511 groups = 7x73
// time chunks -> 3-phase block scan:
//   P1: per (tile,chunk) zero-state response Z_c over 73 groups  (1536 waves)
//   P2: boundary scan S_{c+1} = P·S_c + Z_c,  P = M^73           (256 waves)
//   P3: replay each chunk from its true start, writing outputs   (1792 waves)
// Noise (134MB) < L2 (192MB), so the P3 re-read mostly hits L2; HBM traffic
// stays ~268MB -> ~11.5us floor at 23.3 TB/s.  Falls back to the single-phase
// kernel if ws_size < ~3MB.
// ============================================================================

#define AR_N     16
#define AR_B     4096
#define AR_STEPS 8192
#define AR_G     511
#define AR_CH    7          // time chunks
#define AR_GPC   73         // groups per chunk (7*73 = 511)
#define AR_TILES (AR_B / 16)

typedef float v2f __attribute__((ext_vector_type(2)));
typedef float v4f __attribute__((ext_vector_type(4)));
typedef float v8f __attribute__((ext_vector_type(8)));

// ws float-offset layout
#define WS_M 0
#define WS_L 256
#define WS_P 512
#define WS_Z 1024                          // Z_c, c=0..5: [(c*TILES+tile)*256 + lane*8]
#define WS_S (WS_Z + 6 * AR_TILES * 256)   // S_{c+1}, slot c=0..5
#define WS_TOTAL (WS_S + 6 * AR_TILES * 256)

// ---------------------------------------------------------------------------
// helpers
// ---------------------------------------------------------------------------
__device__ __forceinline__ v8f load_w8(const float* p) {
    v4f lo = *(const v4f*)(p);
    v4f hi = *(const v4f*)(p + 4);
    v8f W;
    W.s0 = lo.x; W.s1 = lo.y; W.s2 = lo.z; W.s3 = lo.w;
    W.s4 = hi.x; W.s5 = hi.y; W.s6 = hi.z; W.s7 = hi.w;
    return W;
}
__device__ __forceinline__ void store_w8(float* p, v8f W) {
    *(v4f*)(p)     = (v4f){W.s0, W.s1, W.s2, W.s3};
    *(v4f*)(p + 4) = (v4f){W.s4, W.s5, W.s6, W.s7};
}
// A-operand slices (A layout: a[kb].{x,y} = Mat[m][4kb+2h+{0,1}], m = lane&15)
__device__ __forceinline__ void load_A4(const float* row, int h, v2f a[4]) {
#pragma unroll
    for (int kb = 0; kb < 4; ++kb) {
        a[kb].x = row[4 * kb + 2 * h + 0];
        a[kb].y = row[4 * kb + 2 * h + 1];
    }
}

// Mat(16x16, A slices) x W(16x16, C/D layout) -> C/D layout, C starts at 0.
// B layout: b[kb].{x,y} = row (4kb+2h+{0,1}), col = lane&15.
__device__ __forceinline__ v8f matW(const v2f a[4], v8f W, int h) {
    float t0 = __shfl_xor(W.s0, 16, 32);
    float t1 = __shfl_xor(W.s1, 16, 32);
    float t2 = __shfl_xor(W.s2, 16, 32);
    float t3 = __shfl_xor(W.s3, 16, 32);
    float t4 = __shfl_xor(W.s4, 16, 32);
    float t5 = __shfl_xor(W.s5, 16, 32);
    float t6 = __shfl_xor(W.s6, 16, 32);
    float t7 = __shfl_xor(W.s7, 16, 32);
    v2f b0, b1, b2, b3;
    b0.x = h ? t2   : W.s0;  b0.y = h ? t3   : W.s1;   // rows 0..3
    b1.x = h ? t6   : W.s4;  b1.y = h ? t7   : W.s5;   // rows 4..7
    b2.x = h ? W.s2 : t0;    b2.y = h ? W.s3 : t1;     // rows 8..11
    b3.x = h ? W.s6 : t4;    b3.y = h ? W.s7 : t5;     // rows 12..15
    v8f acc = {};
    acc = __builtin_amdgcn_wmma_f32_16x16x4_f32(false, a[0], false, b0, (short)0, acc, false, false);
    acc = __builtin_amdgcn_wmma_f32_16x16x4_f32(false, a[1], false, b1, (short)0, acc, false, false);
    acc = __builtin_amdgcn_wmma_f32_16x16x4_f32(false, a[2], false, b2, (short)0, acc, false, false);
    acc = __builtin_amdgcn_wmma_f32_16x16x4_f32(false, a[3], false, b3, (short)0, acc, false, false);
    return acc;
}

// The serial 16-step-group loop.  np = noise + 2h*B + b0 + r (lane base),
// op = out + brow*STEPS + 8h (lane base, only used when STORE).
template <bool STORE>
__device__ __forceinline__ v8f run_groups(v8f W, int g0, int ng,
                                          const v2f aM[4], const v2f aL[4],
                                          const float* np, float* op, int h) {
    np += (size_t)g0 * 16 * AR_B;
    if (STORE) op += 16 * g0;
    for (int g = 0; g < ng; ++g) {
        v2f eb[4];
#pragma unroll
        for (int kb = 0; kb < 4; ++kb) {            // E_g straight into B layout
            eb[kb].x = np[(size_t)(4 * kb + 0) * AR_B];
            eb[kb].y = np[(size_t)(4 * kb + 1) * AR_B];
        }
        v8f accE = {};
#pragma unroll
        for (int kb = 0; kb < 4; ++kb)
            accE = __builtin_amdgcn_wmma_f32_16x16x4_f32(
                false, aL[kb], false, eb[kb], (short)0, accE, false, false);
        v8f accM = matW(aM, W, h);
        W = accE + accM;                             // w_{g+1} = M w_g + L E_g
        if (STORE) {
            op += 16;
            store_w8(op, W);                         // 16 new outputs, contiguous in t
        }
        np += (size_t)16 * AR_B;
    }
    return W;
}

// ---------------------------------------------------------------------------
// Setup: build M = A^16, L, and (if phased) P = M^73 by binary powering in LDS.
// ---------------------------------------------------------------------------
__device__ __forceinline__ void mat_square_lds(float* sM, int k, int kk) {
    float col[16];
#pragma unroll
    for (int i = 0; i < 16; ++i) {
        float acc = 0.f;
#pragma unroll
        for (int q = 0; q < 16; ++q) acc += sM[i * 16 + q] * sM[q * 16 + kk];
        col[i] = acc;
    }
    __syncthreads();
    if (k < 16) {
#pragma unroll
        for (int i = 0; i < 16; ++i) sM[i * 16 + kk] = col[i];
    }
    __syncthreads();
}
__device__ __forceinline__ void mat_apply_lds(const float* sM, float res[16]) {
    float nr[16];
#pragma unroll
    for (int i = 0; i < 16; ++i) {
        float acc = 0.f;
#pragma unroll
        for (int q = 0; q < 16; ++q) acc += sM[i * 16 + q] * res[q];
        nr[i] = acc;
    }
#pragma unroll
    for (int i = 0; i < 16; ++i) res[i] = nr[i];
}

__global__ void ar_build(const float* __restrict__ coeff,
                         const float* __restrict__ log_noise_std,
                         float* __restrict__ ws, int buildP) {
    __shared__ float sM[256];
    const int k  = threadIdx.x;   // 0..31
    const int kk = k & 15;
    float c[16];
#pragma unroll
    for (int i = 0; i < 16; ++i) c[i] = coeff[i];
    const float nstd = expf(log_noise_std[0]);

    // M column kk = A^16 e_kk   (apply companion A: shift, last = coeff.v)
    float v[16];
#pragma unroll
    for (int i = 0; i < 16; ++i) v[i] = (i == kk) ? 1.f : 0.f;
    for (int p = 0; p < 16; ++p) {
        float nv = 0.f;
#pragma unroll
        for (int q = 0; q < 16; ++q) nv += c[q] * v[q];
#pragma unroll
        for (int i = 0; i < 15; ++i) v[i] = v[i + 1];
        v[15] = nv;
    }
    if (k < 16) {
#pragma unroll
        for (int i = 0; i < 16; ++i) { ws[WS_M + i * 16 + kk] = v[i]; sM[i * 16 + kk] = v[i]; }
    }
    // L column kk = nstd * (A^(15-kk))[:,15]
    float w[16];
#pragma unroll
    for (int i = 0; i < 16; ++i) w[i] = (i == 15) ? 1.f : 0.f;
    for (int p = 0; p < 15 - kk; ++p) {
        float nw = 0.f;
#pragma unroll
        for (int q = 0; q < 16; ++q) nw += c[q] * w[q];
#pragma unroll
        for (int i = 0; i < 15; ++i) w[i] = w[i + 1];
        w[15] = nw;
    }
    if (k < 16) {
#pragma unroll
        for (int i = 0; i < 16; ++i) ws[WS_L + i * 16 + kk] = nstd * w[i];
    }
    __syncthreads();

    if (buildP) {
        // res = M^73 e_kk ; 73 = 64 + 8 + 1
        float res[16];
#pragma unroll
        for (int i = 0; i < 16; ++i) res[i] = (i == kk) ? 1.f : 0.f;
        mat_apply_lds(sM, res);                                   // * M
        mat_square_lds(sM, k, kk); mat_square_lds(sM, k, kk); mat_square_lds(sM, k, kk); // sM = M^8
        mat_apply_lds(sM, res);                                   // * M^8
        mat_square_lds(sM, k, kk); mat_square_lds(sM, k, kk); mat_square_lds(sM, k, kk); // sM = M^64
        mat_apply_lds(sM, res);                                   // * M^64
        if (k < 16) {
#pragma unroll
            for (int i = 0; i < 16; ++i) ws[WS_P + i * 16 + kk] = res[i];
        }
    }
}

// ---------------------------------------------------------------------------
// Phase 1: zero-state (chunk 0: true-state) response over 73 groups -> Z_c
// ---------------------------------------------------------------------------
__global__ void __launch_bounds__(32) ar_phase1(const float* __restrict__ init,
                                                const float* __restrict__ noise,
                                                float* __restrict__ ws) {
    const int lane = threadIdx.x, r = lane & 15, h = lane >> 4;
    const int tile = blockIdx.x, c = blockIdx.y;
    const int b0 = tile * 16, brow = b0 + r;

    v2f aM[4], aL[4];
    load_A4(ws + WS_M + r * 16, h, aM);
    load_A4(ws + WS_L + r * 16, h, aL);

    v8f W = {};
    if (c == 0) W = load_w8(init + (size_t)brow * AR_N + 8 * h);

    const float* np = noise + (size_t)(2 * h) * AR_B + (b0 + r);
    W = run_groups<false>(W, c * AR_GPC, AR_GPC, aM, aL, np, nullptr, h);
    store_w8(ws + WS_Z + ((size_t)c * AR_TILES + tile) * 256 + lane * 8, W);
}

// ---------------------------------------------------------------------------
// Phase 2: boundary scan  S_1 = Z_0 ; S_{c+1} = P·S_c + Z_c  (c = 1..5)
// ---------------------------------------------------------------------------
__global__ void __launch_bounds__(32) ar_phase2(float* __restrict__ ws) {
    const int lane = threadIdx.x, r = lane & 15, h = lane >> 4;
    const int tile = blockIdx.x;
    v2f aP[4];
    load_A4(ws + WS_P + r * 16, h, aP);

    v8f W = load_w8(ws + WS_Z + (size_t)tile * 256 + lane * 8);      // Z_0 = S_1
    store_w8(ws + WS_S + (size_t)tile * 256 + lane * 8, W);
    for (int c = 1; c <= 5; ++c) {
        v8f Z = load_w8(ws + WS_Z + ((size_t)c * AR_TILES + tile) * 256 + lane * 8);
        W = matW(aP, W, h) + Z;                                      // S_{c+1}
        store_w8(ws + WS_S + ((size_t)c * AR_TILES + tile) * 256 + lane * 8, W);
    }
}

// ---------------------------------------------------------------------------
// Phase 3: replay each chunk from its true start, writing all outputs.
// ---------------------------------------------------------------------------
__global__ void __launch_bounds__(32) ar_phase3(const float* __restrict__ init,
                                                const float* __restrict__ noise,
                                                const float* __restrict__ ws,
                                                float* __restrict__ out) {
    const int lane = threadIdx.x, r = lane & 15, h = lane >> 4;
    const int tile = blockIdx.x, c = blockIdx.y;
    const int b0 = tile * 16, brow = b0 + r;

    v2f aM[4], aL[4];
    load_A4(ws + WS_M + r * 16, h, aM);
    load_A4(ws + WS_L + r * 16, h, aL);

    v8f W;
    if (c == 0) W = load_w8(init + (size_t)brow * AR_N + 8 * h);
    else        W = load_w8(ws + WS_S + ((size_t)(c - 1) * AR_TILES + tile) * 256 + lane * 8);

    float* op = out + (size_t)brow * AR_STEPS + 8 * h;
    if (c == 0) store_w8(op, W);                       // t = 0..15 (initial values)

    const float* np = noise + (size_t)(2 * h) * AR_B + (b0 + r);
    run_groups<true>(W, c * AR_GPC, AR_GPC, aM, aL, np, op, h);
}

// ---------------------------------------------------------------------------
// Fallback: single-phase full scan (workspace-light path).
// ---------------------------------------------------------------------------
__global__ void __launch_bounds__(32) ar_single(const float* __restrict__ init,
                                                const float* __restrict__ noise,
                                                const float* __restrict__ ws,
                                                float* __restrict__ out) {
    const int lane = threadIdx.x, r = lane & 15, h = lane >> 4;
    const int b0 = blockIdx.x * 16, brow = b0 + r;

    v2f aM[4], aL[4];
    load_A4(ws + WS_M + r * 16, h, aM);
    load_A4(ws + WS_L + r * 16, h, aL);

    v8f W = load_w8(init + (size_t)brow * AR_N + 8 * h);
    float* op = out + (size_t)brow * AR_STEPS + 8 * h;
    store_w8(op, W);
    const float* np = noise + (size_t)(2 * h) * AR_B + (b0 + r);
    run_groups<true>(W, 0, AR_G, aM, aL, np, op, h);
}

// ---------------------------------------------------------------------------
// inputs: 0=initial_values(4096,16) f32, 1=coefficients(16), 2=log_noise_std(1),
//         3=noise(8176,4096), 4=steps (unused)
// ---------------------------------------------------------------------------
extern "C" void kernel_launch(void* const* d_in, const int* in_sizes, int n_in,
                              void* d_out, int out_size, void* d_ws, size_t ws_size,
                              hipStream_t stream) {
    (void)in_sizes; (void)n_in; (void)out_size;
    const float* init  = (const float*)d_in[0];
    const float* coeff = (const float*)d_in[1];
    const float* lstd  = (const float*)d_in[2];
    const float* noise = (const float*)d_in[3];
    float* ws  = (float*)d_ws;
    float* out = (float*)d_out;

    const bool phased = ws_size >= (size_t)WS_TOTAL * sizeof(float);   // ~3 MB
    ar_build<<<1, 32, 0, stream>>>(coeff, lstd, ws, phased ? 1 : 0);
    if (phased) {
        ar_phase1<<<dim3(AR_TILES, AR_CH - 1), 32, 0, stream>>>(init, noise, ws);
        ar_phase2<<<AR_TILES, 32, 0, stream>>>(ws);
        ar_phase3<<<dim3(AR_TILES, AR_CH), 32, 0, stream>>>(init, noise, ws, out);
    } else {
        ar_single<<<AR_TILES, 32, 0, stream>>>(init, noise, ws, out);
    }
}